// PSFDeformableAttention_11020886082044
// MI455X (gfx1250) — compile-verified
//
#include <hip/hip_runtime.h>
#include <hip/hip_bf16.h>

// Problem constants (match setup_inputs): B=4, C=128, H=W=128, K=8, RADIUS=4
#define Bn    4
#define Cn    128
#define Hn    128
#define Wn    128
#define Kn    8
#define HWn   (Hn * Wn)      // 16384
#define QKVC  (3 * Cn)       // 384

typedef float v2f __attribute__((ext_vector_type(2)));
typedef float v8f __attribute__((ext_vector_type(8)));

// ---------------------------------------------------------------------------
// Kernel 1: qkv[b,o,p] = sum_c w_qkv[o,c] * x[b,c,p], stored TRANSPOSED as
// qkv_t[b, p, o]  (pixel-major, channel-contiguous) so every downstream
// per-pixel access is a coalesced 512B burst across the wave.
// One wave computes a 16(M=out-ch) x 32(N=pixel) tile: per K-step one A
// fragment is reused by two WMMAs (two N half-tiles) -> 64 f32 WMMAs/wave.
// ---------------------------------------------------------------------------
__global__ __launch_bounds__(256)
void psf_qkv_gemm(const float* __restrict__ x,
                  const float* __restrict__ w_qkv,
                  float* __restrict__ qkv_t) {
    const int lane = threadIdx.x & 31;
    const int t    = blockIdx.x * 8 + (threadIdx.x >> 5);   // tile id
    const int nt   = t & 511;             // pixel tile (16384/32 = 512)
    const int mb   = t >> 9;              // 0..95 = (b, mtile)
    const int mt   = mb % 24;             // 384/16 = 24 M tiles
    const int b    = mb / 24;
    const int m0   = mt * 16;
    const int n0   = nt * 32;
    const int col  = lane & 15;
    const int hl   = lane >> 4;           // half-wave select

    // A (w_qkv, 16x4 tile): lanes 0-15 carry K = {c0, c0+1}, lanes 16-31 K = {c0+2, c0+3}
    const float* Arow = w_qkv + (size_t)(m0 + col) * Cn;
    // B (x, 4x16 tiles): row K striped across lanes within a VGPR, same half-wave split
    const float* Bc0 = x + (size_t)b * Cn * HWn + (size_t)n0 + col;
    const float* Bc1 = Bc0 + 16;

    v8f acc0 = {};
    v8f acc1 = {};
#pragma unroll 4
    for (int c0 = 0; c0 < Cn; c0 += 4) {
        v2f a = *(const v2f*)(Arow + c0 + 2 * hl);          // 8B contiguous, reused 2x
        v2f b0, b1;
        const size_t r0 = (size_t)(c0 + 2 * hl)     * HWn;
        const size_t r1 = (size_t)(c0 + 2 * hl + 1) * HWn;
        b0.x = Bc0[r0]; b0.y = Bc0[r1];
        b1.x = Bc1[r0]; b1.y = Bc1[r1];
        acc0 = __builtin_amdgcn_wmma_f32_16x16x4_f32(
            false, a, false, b0, (short)0, acc0, false, false);
        acc1 = __builtin_amdgcn_wmma_f32_16x16x4_f32(
            false, a, false, b1, (short)0, acc1, false, false);
    }

    // D layout: VGPR r holds (M = r + 8*hl, N = col).  In the transposed output
    // the 8 M-values per lane are CONSECUTIVE floats -> two b128 stores per tile.
    float* D0 = qkv_t + ((size_t)b * HWn + n0 + col) * QKVC + m0 + 8 * hl;
    float* D1 = D0 + (size_t)16 * QKVC;
    float4 v;
    v.x = acc0[0]; v.y = acc0[1]; v.z = acc0[2]; v.w = acc0[3]; *(float4*)(D0)     = v;
    v.x = acc0[4]; v.y = acc0[5]; v.z = acc0[6]; v.w = acc0[7]; *(float4*)(D0 + 4) = v;
    v.x = acc1[0]; v.y = acc1[1]; v.z = acc1[2]; v.w = acc1[3]; *(float4*)(D1)     = v;
    v.x = acc1[4]; v.y = acc1[5]; v.z = acc1[6]; v.w = acc1[7]; *(float4*)(D1 + 4) = v;
}

// ---------------------------------------------------------------------------
// Kernel 2: deformable sampling + K=8 softmax attention.
// One wave per pixel; lane owns channels [4*lane, 4*lane+4) -> every q/k/v
// access is a wave-coalesced 512B float4 burst from the channel-contiguous
// qkv_t layout.  Logit reduction via shfl_xor (wave32).
// ---------------------------------------------------------------------------
__global__ __launch_bounds__(256)
void psf_attn(const float* __restrict__ qkv_t,
              const int*   __restrict__ psf,
              const float* __restrict__ delta,
              float*       __restrict__ att_t) {
    const int lane = threadIdx.x & 31;
    const int g    = blockIdx.x * 8 + (threadIdx.x >> 5);   // global pixel id
    const int b    = g >> 14;                               // / HWn

    // tanh(delta) * RADIUS  (16 uniform scalars)
    float dx[Kn], dy[Kn];
#pragma unroll
    for (int k = 0; k < Kn; ++k) {
        dx[k] = tanhf(delta[2 * k + 0]) * 4.0f;
        dy[k] = tanhf(delta[2 * k + 1]) * 4.0f;
    }

    const float4 qv = *(const float4*)(qkv_t + (size_t)g * QKVC + 4 * lane);
    const float* kbase = qkv_t + (size_t)b * HWn * QKVC + Cn     + 4 * lane;
    const float* vbase = qkv_t + (size_t)b * HWn * QKVC + 2 * Cn + 4 * lane;

    float  logit[Kn];
    float4 vs[Kn];

#pragma unroll
    for (int k = 0; k < Kn; ++k) {
        const float offx = (float)psf[((size_t)g * Kn + k) * 2 + 0] + dx[k];
        const float offy = (float)psf[((size_t)g * Kn + k) * 2 + 1] + dy[k];
        // grid = offs/[H-1,W-1]*2-1 ; ix=(g0+1)/2*(W-1) ; iy=(g1+1)/2*(H-1)
        const float ix = offx * ((float)(Wn - 1) / (float)(Hn - 1));
        const float iy = offy * ((float)(Hn - 1) / (float)(Wn - 1));
        const float x0 = floorf(ix), y0 = floorf(iy);
        const float wx1 = ix - x0, wx0 = 1.0f - wx1;
        const float wy1 = iy - y0, wy0 = 1.0f - wy1;

        float4 ks; ks.x = ks.y = ks.z = ks.w = 0.0f;
        float4 vv; vv.x = vv.y = vv.z = vv.w = 0.0f;
#pragma unroll
        for (int tap = 0; tap < 4; ++tap) {
            const float xf = (tap & 1) ? x0 + 1.0f : x0;
            const float yf = (tap & 2) ? y0 + 1.0f : y0;
            const float wgt = ((tap & 1) ? wx1 : wx0) * ((tap & 2) ? wy1 : wy0);
            const bool valid = (xf >= 0.0f) && (xf <= (float)(Wn - 1)) &&
                               (yf >= 0.0f) && (yf <= (float)(Hn - 1));
            const int xi = (int)fminf(fmaxf(xf, 0.0f), (float)(Wn - 1));
            const int yi = (int)fminf(fmaxf(yf, 0.0f), (float)(Hn - 1));
            const float wv = valid ? wgt : 0.0f;
            const size_t po = (size_t)(yi * Wn + xi) * QKVC;
            const float4 kt = *(const float4*)(kbase + po);
            const float4 vt = *(const float4*)(vbase + po);
            ks.x += wv * kt.x; ks.y += wv * kt.y; ks.z += wv * kt.z; ks.w += wv * kt.w;
            vv.x += wv * vt.x; vv.y += wv * vt.y; vv.z += wv * vt.z; vv.w += wv * vt.w;
        }
        vs[k]    = vv;
        logit[k] = qv.x * ks.x + qv.y * ks.y + qv.z * ks.z + qv.w * ks.w;
    }

    // wave32 all-reduce of each logit, then scale by C^-1/2
#pragma unroll
    for (int k = 0; k < Kn; ++k) {
        float r = logit[k];
#pragma unroll
        for (int off = 16; off > 0; off >>= 1) r += __shfl_xor(r, off, 32);
        logit[k] = r * 0.08838834764831843f;   // 128^-0.5
    }

    // softmax over K=8 (uniform across lanes after reduction)
    float mx = logit[0];
#pragma unroll
    for (int k = 1; k < Kn; ++k) mx = fmaxf(mx, logit[k]);
    float p[Kn]; float s = 0.0f;
#pragma unroll
    for (int k = 0; k < Kn; ++k) { p[k] = expf(logit[k] - mx); s += p[k]; }
    const float inv = 1.0f / s;

    float4 o; o.x = o.y = o.z = o.w = 0.0f;
#pragma unroll
    for (int k = 0; k < Kn; ++k) {
        const float w = p[k] * inv;
        o.x += w * vs[k].x; o.y += w * vs[k].y; o.z += w * vs[k].z; o.w += w * vs[k].w;
    }
    *(float4*)(att_t + (size_t)g * Cn + 4 * lane) = o;   // (B, HW, C), coalesced
}

// ---------------------------------------------------------------------------
// Kernel 3: out[b,o,p] = x[b,o,p] + sum_c w_proj[o,c] * att[b,c,p]
// att_t is channel-contiguous -> B-matrix fragments are contiguous v2f loads.
// Same 16x32 double-N tile as kernel 1 (A fragment reused by two WMMAs).
// ---------------------------------------------------------------------------
__global__ __launch_bounds__(256)
void psf_proj_gemm(const float* __restrict__ att_t,
                   const float* __restrict__ w_proj,
                   const float* __restrict__ x,
                   float* __restrict__ out) {
    const int lane = threadIdx.x & 31;
    const int t    = blockIdx.x * 8 + (threadIdx.x >> 5);
    const int nt   = t & 511;            // 512 N tiles of 32 pixels
    const int mb   = t >> 9;             // 0..31
    const int mt   = mb & 7;             // 128/16 = 8 M tiles
    const int b    = mb >> 3;
    const int m0   = mt * 16;
    const int n0   = nt * 32;
    const int col  = lane & 15;
    const int hl   = lane >> 4;

    const float* Arow = w_proj + (size_t)(m0 + col) * Cn;
    const float* Bc0  = att_t + ((size_t)b * HWn + n0 + col) * Cn;
    const float* Bc1  = Bc0 + (size_t)16 * Cn;

    v8f acc0 = {};
    v8f acc1 = {};
#pragma unroll 4
    for (int c0 = 0; c0 < Cn; c0 += 4) {
        v2f a  = *(const v2f*)(Arow + c0 + 2 * hl);
        v2f b0 = *(const v2f*)(Bc0  + c0 + 2 * hl);   // contiguous in att_t
        v2f b1 = *(const v2f*)(Bc1  + c0 + 2 * hl);
        acc0 = __builtin_amdgcn_wmma_f32_16x16x4_f32(
            false, a, false, b0, (short)0, acc0, false, false);
        acc1 = __builtin_amdgcn_wmma_f32_16x16x4_f32(
            false, a, false, b1, (short)0, acc1, false, false);
    }

    // fused residual + store to (B, C, HW)
    const size_t base = (size_t)b * Cn * HWn + (size_t)n0 + col;
#pragma unroll
    for (int r = 0; r < 8; ++r) {
        const int m = m0 + r + 8 * hl;
        const size_t idx = base + (size_t)m * HWn;
        out[idx]      = x[idx]      + acc0[r];
        out[idx + 16] = x[idx + 16] + acc1[r];
    }
}

// ---------------------------------------------------------------------------
extern "C" void kernel_launch(void* const* d_in, const int* in_sizes, int n_in,
                              void* d_out, int out_size, void* d_ws, size_t ws_size,
                              hipStream_t stream) {
    const float* x      = (const float*)d_in[0];   // (4,128,128,128)
    const int*   psf    = (const int*)  d_in[1];   // (4,128,128,8,2)
    const float* delta  = (const float*)d_in[2];   // (1,1,1,8,2)
    const float* w_qkv  = (const float*)d_in[3];   // (384,128)
    const float* w_proj = (const float*)d_in[4];   // (128,128)
    float*       out    = (float*)d_out;           // (4,128,128,128)

    float* qkv_t = (float*)d_ws;                             // B*HW*384 = 25,165,824 f32
    float* att_t = qkv_t + (size_t)Bn * HWn * QKVC;          // B*HW*128 =  8,388,608 f32

    // K1: 4 batches * 24 Mtiles * 512 Ntiles = 49152 waves / 8 per block
    psf_qkv_gemm<<<6144, 256, 0, stream>>>(x, w_qkv, qkv_t);
    // K2: 65536 pixels, one wave each, 8 waves per block
    psf_attn<<<8192, 256, 0, stream>>>(qkv_t, psf, delta, att_t);
    // K3: 4 * 8 * 512 = 16384 waves / 8 per block
    psf_proj_gemm<<<2048, 256, 0, stream>>>(att_t, w_proj, x, out);
}